// _SNRLoss_67525475827756
// MI455X (gfx1250) — compile-verified
//
#include <hip/hip_runtime.h>
#include <stdint.h>

// SNR loss: per-row 8192-pt rFFT (4096-pt complex FFT, 3 Stockham radix-16
// stages on V_WMMA_F32_16X16X4_F32), fused power-spectrum weighting and
// reduction. One workgroup (256 threads = 8 waves) per row; 32KB LDS.
// Global->LDS staging uses GLOBAL_LOAD_ASYNC_TO_LDS_B128 (ASYNCcnt path).
// Twiddles use incremental complex rotations (1-2 sincos per group instead of
// one per element) to keep VALU well under the 23.3 TB/s HBM roofline.

typedef float v2f __attribute__((ext_vector_type(2)));
typedef float v8f __attribute__((ext_vector_type(8)));

#define PI_F 3.14159265358979323846f

__device__ __forceinline__ v8f wmma4(v2f a, v2f b, v8f c) {
  // (neg_a, A, neg_b, B, c_mod, C, reuse_a, reuse_b)
  return __builtin_amdgcn_wmma_f32_16x16x4_f32(false, a, false, b, (short)0, c,
                                               false, false);
}

__global__ __launch_bounds__(256) void snr_row_kernel(
    const float* __restrict__ x, const float* __restrict__ targets,
    const int* __restrict__ fsp, float* __restrict__ row_loss) {
  __shared__ float2 z[4096];  // 32 KB

  const int tid  = threadIdx.x;
  const int row  = blockIdx.x;
  const int lane = tid & 31;
  const int wv   = tid >> 5;     // wave id 0..7
  const int col  = lane & 15;    // N (column) for B/C/D, M (row) for A
  const int half = lane >> 4;    // lane group
  const int rof  = half << 1;    // K offset inside a K=4 block (0 or 2)

  // ---- async copy 8192 reals -> LDS (identity bytes: z[k]=(x[2k],x[2k+1]))
  {
    const uint64_t gbase = (uint64_t)(uintptr_t)(x + (size_t)row * 8192);
    const uint32_t lbase = (uint32_t)(uintptr_t)(&z[0]);
#pragma unroll
    for (int i = 0; i < 8; ++i) {
      uint32_t u = (uint32_t)(i * 256 + tid);      // 16-byte chunk index
      uint64_t gaddr = gbase + (uint64_t)u * 16u;
      uint32_t laddr = lbase + u * 16u;
      asm volatile("global_load_async_to_lds_b128 %0, %1, off"
                   :: "v"(laddr), "v"(gaddr)
                   : "memory");
    }
  }

  // ---- DFT16 matrix fragments: A[k][r] = exp(-i*2*pi*k*r/16) ----
  // f32 16x4 A layout: lane M = lane&15; VGPR0 holds K = rof, VGPR1 K = rof+1;
  // chained j covers K = 4j..4j+3.
  v2f Ar[4], Ai[4], nAi[4];
  const float kf = (float)col;
#pragma unroll
  for (int j = 0; j < 4; ++j) {
    float r0 = (float)(4 * j + rof);
    float s0, c0, s1, c1;
    __sincosf(-(PI_F / 8.0f) * kf * r0, &s0, &c0);
    __sincosf(-(PI_F / 8.0f) * kf * (r0 + 1.0f), &s1, &c1);
    Ar[j]  = (v2f){c0, c1};
    Ai[j]  = (v2f){s0, s1};
    nAi[j] = (v2f){-s0, -s1};
  }

  asm volatile("s_wait_asynccnt 0x0" ::: "memory");  // LDS tile resident
  __syncthreads();

  // ---- 3 Stockham radix-16 stages (n = 4096 -> 256 -> 16), in-place:
  // all waves preload their B tiles, barrier, matmul+twiddle, scatter, barrier.
#pragma unroll
  for (int st = 0; st < 3; ++st) {
    v2f Br[2][4], Bi[2][4];
#pragma unroll
    for (int tt = 0; tt < 2; ++tt) {
      int T = wv * 2 + tt;       // tile 0..15
#pragma unroll
      for (int j = 0; j < 4; ++j) {
        int r0 = 4 * j + rof;
        int i0;
        if (st == 1) i0 = T + 16 * col + 256 * r0;       // q=T, p=col
        else         i0 = (16 * T + col) + 256 * r0;     // p (st0) / q (st2)
        float2 e0 = z[i0];
        float2 e1 = z[i0 + 256];
        Br[tt][j] = (v2f){e0.x, e1.x};
        Bi[tt][j] = (v2f){e0.y, e1.y};
      }
    }
    __syncthreads();
#pragma unroll
    for (int tt = 0; tt < 2; ++tt) {
      int T = wv * 2 + tt;
      v8f Dr = {0, 0, 0, 0, 0, 0, 0, 0};
      v8f Di = {0, 0, 0, 0, 0, 0, 0, 0};
      // complex D = A * B : Dr = Ar*Br - Ai*Bi ; Di = Ar*Bi + Ai*Br
#pragma unroll
      for (int j = 0; j < 4; ++j) Dr = wmma4(Ar[j], Br[tt][j], Dr);
#pragma unroll
      for (int j = 0; j < 4; ++j) Dr = wmma4(nAi[j], Bi[tt][j], Dr);
#pragma unroll
      for (int j = 0; j < 4; ++j) Di = wmma4(Ar[j], Bi[tt][j], Di);
#pragma unroll
      for (int j = 0; j < 4; ++j) Di = wmma4(Ai[j], Br[tt][j], Di);

      // D layout: VGPR v -> M (=k) = v + 8*half, N = col.
      // Twiddle w_n^{p*k}: constant angle step d = -2*pi*p/n as k advances,
      // so 2 sincos + 8 incremental rotations per tile.
      const int p0 = (st == 0) ? (16 * T + col) : col;
      float cs = 1.0f, ss = 0.0f, cr = 1.0f, sr = 0.0f;
      if (st == 0) {
        float d = -2.0f * PI_F * (float)p0 * (1.0f / 4096.0f);
        __sincosf(d, &ss, &cs);
        __sincosf(d * (float)(8 * half), &sr, &cr);
      } else if (st == 1) {
        float d = -2.0f * PI_F * (float)p0 * (1.0f / 256.0f);
        __sincosf(d, &ss, &cs);
        __sincosf(d * (float)(8 * half), &sr, &cr);
      }
#pragma unroll
      for (int v = 0; v < 8; ++v) {
        int k = v + half * 8;
        float re = Dr[v], im = Di[v];
        int oidx;
        if (st == 0) {
          oidx = 16 * p0 + k;
        } else if (st == 1) {
          oidx = T + 256 * p0 + 16 * k;
        } else {
          oidx = (16 * T + col) + 256 * k;  // last stage: no twiddle
        }
        if (st < 2) {
          float nr = re * cr - im * sr;
          im = re * sr + im * cr;
          re = nr;
          float nc = cr * cs - sr * ss;      // advance rotation by d
          sr = cr * ss + sr * cs;
          cr = nc;
        }
        z[oidx] = make_float2(re, im);
      }
    }
    __syncthreads();
  }

  // ---- real-FFT unpack + weighted power sums (fused, no P1 storage) ----
  const float tgt    = targets[row];
  const float halfFs = 0.5f * (float)fsp[0];
  const float scale  = 4096.0f / halfFs;  // bins per Hz
  const int   r    = (int)floorf(tgt * scale + 0.5f);
  const int   minI = (int)floorf((40.0f / 60.0f) * scale + 0.5f);
  const int   maxI = (int)floorf((250.0f / 60.0f) * scale + 0.5f);

  // Unpack twiddle e^{-i*pi*i/4096}: step per iteration (i += 256) is the
  // constant rotation by pi/16 -> 1 sincos + incremental rotation.
  float cth, sth;
  __sincosf(PI_F * (float)tid * (1.0f / 4096.0f), &sth, &cth);
  const float cS = 0.980785280403230449f;  // cos(pi/16)
  const float sS = 0.195090322016128268f;  // sin(pi/16)

  float sig = 0.0f, noi = 0.0f;
  for (int i = tid; i <= 4096; i += 256) {
    float2 a  = z[i & 4095];
    float2 bb = z[(4096 - i) & 4095];
    // Xe = (Z[k]+conj(Z[M-k]))/2 ; Xo = (Z[k]-conj(Z[M-k]))/2
    float er = 0.5f * (a.x + bb.x);
    float ei = 0.5f * (a.y - bb.y);
    float pr = 0.5f * (a.x - bb.x);
    float pi = 0.5f * (a.y + bb.y);
    float tr = cth, ti = -sth;
    // X = Xe + (ti - i*tr) * Xo
    float Xr = er + ti * pr + tr * pi;
    float Xi = ei + ti * pi - tr * pr;
    float P = (Xr * Xr + Xi * Xi) * (1.0f / 8192.0f);  // |rfft|^2 / n
    float u = (i == r - 1 || i == r + 1)
                  ? 0.5f
                  : ((i == r || i == 2 * r) ? 1.0f : 0.0f);
    bool ex = (i < minI) || (i >= maxI) || (i == 2 * r - 1) || (i == 2 * r + 1);
    float w = ex ? 0.0f : (1.0f - u);
    sig += P * u;
    noi += P * w;
    float nc = cth * cS - sth * sS;  // advance by pi/16
    sth = cth * sS + sth * cS;
    cth = nc;
  }

  __syncthreads();                       // all reads of z done
  z[tid] = make_float2(sig, noi);        // reuse LDS for block reduction
  __syncthreads();
  for (int off = 128; off > 0; off >>= 1) {
    if (tid < off) {
      z[tid].x += z[tid + off].x;
      z[tid].y += z[tid + off].y;
    }
    __syncthreads();
  }
  if (tid == 0) {
    row_loss[row] = -10.0f * log10f(z[0].x / (z[0].y + 1.0f));
  }
}

__global__ __launch_bounds__(256) void reduce_mean_kernel(
    const float* __restrict__ rl, float* __restrict__ out, int n) {
  __shared__ float red[256];
  float s = 0.0f;
  for (int i = threadIdx.x; i < n; i += 256) s += rl[i];
  red[threadIdx.x] = s;
  __syncthreads();
  for (int off = 128; off > 0; off >>= 1) {
    if (threadIdx.x < off) red[threadIdx.x] += red[threadIdx.x + off];
    __syncthreads();
  }
  if (threadIdx.x == 0) out[0] = red[0] / (float)n;
}

extern "C" void kernel_launch(void* const* d_in, const int* in_sizes, int n_in,
                              void* d_out, int out_size, void* d_ws,
                              size_t ws_size, hipStream_t stream) {
  const float* outputs = (const float*)d_in[0];
  const float* targets = (const float*)d_in[1];
  const int*   fs      = (const int*)d_in[2];
  (void)n_in; (void)out_size; (void)ws_size;

  const int B = in_sizes[0] / 8192;  // rows (N fixed at 8192 in this kernel)
  float* row_loss = (float*)d_ws;

  snr_row_kernel<<<B, 256, 0, stream>>>(outputs, targets, fs, row_loss);
  reduce_mean_kernel<<<1, 256, 0, stream>>>(row_loss, (float*)d_out, B);
}